// _PointnetSAModuleBase_55327768708591
// MI455X (gfx1250) — compile-verified
//
#include <hip/hip_runtime.h>
#include <hip/hip_bf16.h>

typedef __attribute__((ext_vector_type(2))) float v2f;
typedef __attribute__((ext_vector_type(8))) float v8f;

#define BB 8
#define NN 16384
#define PP 1024
#define SS 32
#define CIN 64
#define K1 67          // 3 + CIN
#define K1P 68         // padded to multiple of 4
#define R2 0.16f       // RADIUS^2

static __device__ __forceinline__ v8f wmma4(v2f a, v2f b, v8f c) {
    // D = A(16x4 f32) * B(4x16 f32) + C(16x16 f32)
    return __builtin_amdgcn_wmma_f32_16x16x4_f32(
        false, a, false, b, (short)0, c, false, false);
}

// ---------------------------------------------------------------------------
// Kernel 1: furthest point sampling. One block per batch, 1024 threads.
// Each thread keeps its 16 points (x,y,z) and running min-dist in REGISTERS,
// so the 1023 serial rounds do no global reads: shuffle + LDS argmax only.
// Writes fps indices (ws) and new_xyz (first section of d_out).
// ---------------------------------------------------------------------------
__global__ void fps_kernel(const float* __restrict__ xyz,
                           float* __restrict__ new_xyz,
                           int* __restrict__ fps_idx) {
    const int b    = blockIdx.x;
    const int tid  = threadIdx.x;
    const int lane = tid & 31;
    const int wid  = tid >> 5;
    constexpr int PT = NN / 1024;   // 16 points per thread

    float px[PT], py[PT], pz[PT], md[PT];
    const float* xb = xyz + (size_t)b * NN * 3;
#pragma unroll
    for (int k = 0; k < PT; ++k) {
        int i = tid + k * 1024;
        px[k] = xb[i * 3 + 0];
        py[k] = xb[i * 3 + 1];
        pz[k] = xb[i * 3 + 2];
        md[k] = 3.4e38f;
    }

    __shared__ float sV[32], sX[32], sY[32], sZ[32];
    __shared__ int   sI[32];
    __shared__ float bXs, bYs, bZs;

    if (tid == 0) {
        bXs = xb[0]; bYs = xb[1]; bZs = xb[2];
        fps_idx[b * PP] = 0;
        new_xyz[(size_t)(b * PP) * 3 + 0] = bXs;
        new_xyz[(size_t)(b * PP) * 3 + 1] = bYs;
        new_xyz[(size_t)(b * PP) * 3 + 2] = bZs;
    }
    __syncthreads();

    for (int j = 1; j < PP; ++j) {
        const float lx = bXs, ly = bYs, lz = bZs;
        float bestV = -1.0f, bx = 0.f, by = 0.f, bz = 0.f;
        int   bestI = 0;
#pragma unroll
        for (int k = 0; k < PT; ++k) {
            float dx = px[k] - lx, dy = py[k] - ly, dz = pz[k] - lz;
            float d = dx * dx + dy * dy + dz * dz;
            md[k] = fminf(md[k], d);
            if (md[k] > bestV) {
                bestV = md[k]; bestI = tid + k * 1024;
                bx = px[k]; by = py[k]; bz = pz[k];
            }
        }
        // wave32 argmax reduction (prefer smaller index on ties, like argmax)
#pragma unroll
        for (int off = 16; off > 0; off >>= 1) {
            float ov = __shfl_xor(bestV, off, 32);
            int   oi = __shfl_xor(bestI, off, 32);
            float ox = __shfl_xor(bx, off, 32);
            float oy = __shfl_xor(by, off, 32);
            float oz = __shfl_xor(bz, off, 32);
            if (ov > bestV || (ov == bestV && oi < bestI)) {
                bestV = ov; bestI = oi; bx = ox; by = oy; bz = oz;
            }
        }
        if (lane == 0) { sV[wid] = bestV; sI[wid] = bestI; sX[wid] = bx; sY[wid] = by; sZ[wid] = bz; }
        __syncthreads();
        if (wid == 0) {
            bestV = sV[lane]; bestI = sI[lane]; bx = sX[lane]; by = sY[lane]; bz = sZ[lane];
#pragma unroll
            for (int off = 16; off > 0; off >>= 1) {
                float ov = __shfl_xor(bestV, off, 32);
                int   oi = __shfl_xor(bestI, off, 32);
                float ox = __shfl_xor(bx, off, 32);
                float oy = __shfl_xor(by, off, 32);
                float oz = __shfl_xor(bz, off, 32);
                if (ov > bestV || (ov == bestV && oi < bestI)) {
                    bestV = ov; bestI = oi; bx = ox; by = oy; bz = oz;
                }
            }
            if (lane == 0) {
                bXs = bx; bYs = by; bZs = bz;
                fps_idx[b * PP + j] = bestI;
                new_xyz[(size_t)(b * PP + j) * 3 + 0] = bx;
                new_xyz[(size_t)(b * PP + j) * 3 + 1] = by;
                new_xyz[(size_t)(b * PP + j) * 3 + 2] = bz;
            }
        }
        __syncthreads();
    }
}

// ---------------------------------------------------------------------------
// Kernel 2: ball query. One thread per centroid; whole-batch xyz (196 KB) is
// L2-resident and all threads of a block stream it in lockstep (broadcast).
// First 32 in-radius indices ascending, short balls padded with the first hit.
// ---------------------------------------------------------------------------
__global__ void ball_query_kernel(const float* __restrict__ xyz,
                                  const float* __restrict__ new_xyz,
                                  int* __restrict__ gidx) {
    const int cp = blockIdx.x * blockDim.x + threadIdx.x;   // 0 .. B*P-1
    if (cp >= BB * PP) return;
    const int b = cp >> 10;
    const float cx = new_xyz[cp * 3 + 0];
    const float cy = new_xyz[cp * 3 + 1];
    const float cz = new_xyz[cp * 3 + 2];
    const float* xb = xyz + (size_t)b * NN * 3;
    int* gp = gidx + (size_t)cp * SS;

    int cnt = 0, first = 0;
    for (int i = 0; i < NN; ++i) {
        float dx = xb[i * 3 + 0] - cx;
        float dy = xb[i * 3 + 1] - cy;
        float dz = xb[i * 3 + 2] - cz;
        float d2 = dx * dx + dy * dy + dz * dz;
        if (d2 <= R2) {
            if (cnt == 0) first = i;
            gp[cnt] = i;
            if (++cnt == SS) break;
        }
    }
    for (int k = cnt; k < SS; ++k) gp[k] = first;
}

// ---------------------------------------------------------------------------
// Kernel 3: gather + 3-layer shared MLP (fp32 WMMA 16x16x4) + max-pool.
// Block = 128 threads (4 waves). One wave per centroid: M=32 samples ->
// 2 M-tiles. Weights staged once per block in LDS; per-wave 32xK activation
// tile in LDS so A-fragments and the transposed inter-layer handoff are
// simple LDS reads. EXEC is uniform (all 1s) at every WMMA.
// ---------------------------------------------------------------------------
__global__ void group_mlp_kernel(const float* __restrict__ xyz,
                                 const float* __restrict__ features,
                                 const float* __restrict__ W1, const float* __restrict__ b1,
                                 const float* __restrict__ W2, const float* __restrict__ b2,
                                 const float* __restrict__ W3, const float* __restrict__ b3,
                                 const float* __restrict__ new_xyz,
                                 const int* __restrict__ gidx,
                                 float* __restrict__ out_feat) {
    __shared__ float W1s[64][K1P];    // B-matrix source: B[k][n] = W1[n][k]
    __shared__ float W2s[64][64];
    __shared__ float W3s[128][64];
    __shared__ float b1s[64], b2s[64], b3s[128];
    __shared__ float hs[4][SS][K1P];  // per-wave activation tile (rows x K)

    const int tid  = threadIdx.x;
    const int lane = tid & 31;
    const int w    = tid >> 5;
    const int l15  = lane & 15;
    const int kh   = (lane >> 4) << 1;     // lanes 16-31 hold K+2,K+3
    const int rh   = (lane >> 4) << 3;     // lanes 16-31 hold rows M+8 (C/D)

    // ---- stage weights & biases ----
    for (int i = tid; i < 64 * K1; i += 128) W1s[i / K1][i % K1] = W1[i];
    for (int i = tid; i < 64; i += 128)      W1s[i][K1] = 0.0f;       // K pad
    for (int i = tid; i < 64 * 64; i += 128) W2s[i >> 6][i & 63] = W2[i];
    for (int i = tid; i < 128 * 64; i += 128) W3s[i >> 6][i & 63] = W3[i];
    for (int i = tid; i < 64; i += 128) { b1s[i] = b1[i]; b2s[i] = b2[i]; }
    for (int i = tid; i < 128; i += 128) b3s[i] = b3[i];

    // ---- gather inputs for this wave's centroid ----
    const int cp = blockIdx.x * 4 + w;          // centroid id, 0..B*P-1
    const int b  = cp >> 10;
    const int p  = cp & 1023;
    const int g  = gidx[(size_t)cp * SS + lane];     // one sample per lane
    const float cx = new_xyz[cp * 3 + 0];
    const float cy = new_xyz[cp * 3 + 1];
    const float cz = new_xyz[cp * 3 + 2];
    hs[w][lane][0] = xyz[((size_t)b * NN + g) * 3 + 0] - cx;
    hs[w][lane][1] = xyz[((size_t)b * NN + g) * 3 + 1] - cy;
    hs[w][lane][2] = xyz[((size_t)b * NN + g) * 3 + 2] - cz;
    const float* fb = features + (size_t)b * CIN * NN;
#pragma unroll 4
    for (int c = 0; c < CIN; ++c)
        hs[w][lane][3 + c] = fb[(size_t)c * NN + g];
    hs[w][lane][K1] = 0.0f;
    __syncthreads();

    // ---- layer 1: (32 x 68) * (68 x 64) ----
    v8f acc1[2][4];
#pragma unroll
    for (int mt = 0; mt < 2; ++mt)
#pragma unroll
        for (int nt = 0; nt < 4; ++nt) {
            const float bb = b1s[nt * 16 + l15];
            v8f c;
#pragma unroll
            for (int j = 0; j < 8; ++j) c[j] = bb;
#pragma unroll
            for (int k = 0; k < K1P; k += 4) {
                v2f a, bv;
                a.x  = hs[w][mt * 16 + l15][k + kh];
                a.y  = hs[w][mt * 16 + l15][k + kh + 1];
                bv.x = W1s[nt * 16 + l15][k + kh];
                bv.y = W1s[nt * 16 + l15][k + kh + 1];
                c = wmma4(a, bv, c);
            }
#pragma unroll
            for (int j = 0; j < 8; ++j) c[j] = fmaxf(c[j], 0.0f);
            acc1[mt][nt] = c;
        }
    __syncthreads();
    // transposed write-back (C/D layout -> row-major LDS tile)
#pragma unroll
    for (int mt = 0; mt < 2; ++mt)
#pragma unroll
        for (int nt = 0; nt < 4; ++nt) {
            const int col = nt * 16 + l15;
            const int rb  = mt * 16 + rh;
#pragma unroll
            for (int v = 0; v < 8; ++v) hs[w][rb + v][col] = acc1[mt][nt][v];
        }
    __syncthreads();

    // ---- layer 2: (32 x 64) * (64 x 64) ----
    v8f acc2[2][4];
#pragma unroll
    for (int mt = 0; mt < 2; ++mt)
#pragma unroll
        for (int nt = 0; nt < 4; ++nt) {
            const float bb = b2s[nt * 16 + l15];
            v8f c;
#pragma unroll
            for (int j = 0; j < 8; ++j) c[j] = bb;
#pragma unroll
            for (int k = 0; k < 64; k += 4) {
                v2f a, bv;
                a.x  = hs[w][mt * 16 + l15][k + kh];
                a.y  = hs[w][mt * 16 + l15][k + kh + 1];
                bv.x = W2s[nt * 16 + l15][k + kh];
                bv.y = W2s[nt * 16 + l15][k + kh + 1];
                c = wmma4(a, bv, c);
            }
#pragma unroll
            for (int j = 0; j < 8; ++j) c[j] = fmaxf(c[j], 0.0f);
            acc2[mt][nt] = c;
        }
    __syncthreads();
#pragma unroll
    for (int mt = 0; mt < 2; ++mt)
#pragma unroll
        for (int nt = 0; nt < 4; ++nt) {
            const int col = nt * 16 + l15;
            const int rb  = mt * 16 + rh;
#pragma unroll
            for (int v = 0; v < 8; ++v) hs[w][rb + v][col] = acc2[mt][nt][v];
        }
    __syncthreads();

    // ---- layer 3: (32 x 64) * (64 x 128), fused ReLU + max-pool over 32 ----
#pragma unroll
    for (int nt = 0; nt < 8; ++nt) {
        const float bb = b3s[nt * 16 + l15];
        v8f c0, c1;
#pragma unroll
        for (int j = 0; j < 8; ++j) { c0[j] = bb; c1[j] = bb; }
#pragma unroll
        for (int k = 0; k < 64; k += 4) {
            v2f a0, a1, bv;
            a0.x = hs[w][l15][k + kh];
            a0.y = hs[w][l15][k + kh + 1];
            a1.x = hs[w][16 + l15][k + kh];
            a1.y = hs[w][16 + l15][k + kh + 1];
            bv.x = W3s[nt * 16 + l15][k + kh];
            bv.y = W3s[nt * 16 + l15][k + kh + 1];
            c0 = wmma4(a0, bv, c0);
            c1 = wmma4(a1, bv, c1);
        }
        // ReLU + per-lane max over the 16 rows this lane holds
        float m = 0.0f;   // ReLU floor
#pragma unroll
        for (int j = 0; j < 8; ++j) {
            m = fmaxf(m, c0[j]);
            m = fmaxf(m, c1[j]);
        }
        // combine the two half-tiles of each column (lane c <-> lane c+16)
        m = fmaxf(m, __shfl_xor(m, 16, 32));
        if (lane < 16)
            out_feat[((size_t)b * 128 + (nt * 16 + l15)) * PP + p] = m;
    }
}

// ---------------------------------------------------------------------------
extern "C" void kernel_launch(void* const* d_in, const int* in_sizes, int n_in,
                              void* d_out, int out_size, void* d_ws, size_t ws_size,
                              hipStream_t stream) {
    const float* xyz      = (const float*)d_in[0];   // (B,N,3)
    const float* features = (const float*)d_in[1];   // (B,64,N)
    const float* W1 = (const float*)d_in[2];         // (64,67)
    const float* b1 = (const float*)d_in[3];
    const float* W2 = (const float*)d_in[4];         // (64,64)
    const float* b2 = (const float*)d_in[5];
    const float* W3 = (const float*)d_in[6];         // (128,64)
    const float* b3 = (const float*)d_in[7];

    float* new_xyz  = (float*)d_out;                 // (B,P,3)
    float* out_feat = new_xyz + (size_t)BB * PP * 3; // (B,128,P)

    int* fps_idx = (int*)d_ws;                       // B*P ints
    int* gidx    = fps_idx + BB * PP;                // B*P*S ints

    fps_kernel<<<BB, 1024, 0, stream>>>(xyz, new_xyz, fps_idx);
    ball_query_kernel<<<(BB * PP) / 256, 256, 0, stream>>>(xyz, new_xyz, gidx);
    group_mlp_kernel<<<(BB * PP) / 4, 128, 0, stream>>>(
        xyz, features, W1, b1, W2, b2, W3, b3, new_xyz, gidx, out_feat);
}